// GATv2Actor_83313775607886
// MI455X (gfx1250) — compile-verified
//
#include <hip/hip_runtime.h>
#include <hip/hip_bf16.h>
#include <math.h>

#define NN 50000
#define NE 800000
#define ET (NE + NN)            // edges + self loops
#define F  128                  // IN_DIM
#define PW 384                  // node projection width: Asrc(128) Adst(128) V(128)

typedef float v2f __attribute__((ext_vector_type(2)));
typedef float v8f __attribute__((ext_vector_type(8)));

__device__ __forceinline__ float lrelu(float x) { return x > 0.f ? x : 0.2f * x; }

__device__ __forceinline__ void atomicMaxF(float* addr, float v) {
    if (v >= 0.f) atomicMax((int*)addr, __float_as_int(v));
    else          atomicMin((unsigned int*)addr, __float_as_uint(v));
}

// ---------------- init: seg_max=-inf, seg_sum=0, agg=0 ----------------
__global__ void init_ws(float* __restrict__ seg_max, float* __restrict__ seg_sum,
                        float* __restrict__ agg) {
    int i = blockIdx.x * 256 + threadIdx.x;
    if (i < 2 * NN) { seg_max[i] = -INFINITY; seg_sum[i] = 0.f; }
    if (i < F * NN) agg[i] = 0.f;
}

// ---- repack weights: Wt1[j][k] = concat(pairW_src, pairW_dst, valueW)^T,
//      bias1 = pair_b folded onto Adst columns, Wt2 = out_W^T ----
__global__ void build_weights(const float* __restrict__ pair_W, const float* __restrict__ pair_b,
                              const float* __restrict__ value_W, const float* __restrict__ out_W,
                              float* __restrict__ Wt1, float* __restrict__ bias1,
                              float* __restrict__ Wt2) {
    int i = blockIdx.x * 256 + threadIdx.x;
    if (i < PW * F) {
        int j = i >> 7, k = i & 127;           // output col j, input dim k
        float v;
        if (j < 128)      { int h = j >> 6;        v = pair_W[h * 16384 + k * 64 + (j & 63)]; }
        else if (j < 256) { int jj = j - 128; int h = jj >> 6;
                            v = pair_W[h * 16384 + (128 + k) * 64 + (jj & 63)]; }
        else              { int jj = j - 256; int h = jj >> 6;
                            v = value_W[h * 8192 + k * 64 + (jj & 63)]; }
        Wt1[i] = v;
    } else if (i < PW * F + PW) {
        int j = i - PW * F;
        bias1[j] = (j >= 128 && j < 256) ? pair_b[j - 128] : 0.f;   // bias on Adst only
    } else if (i < PW * F + PW + F * F) {
        int t = i - (PW * F + PW);
        int j = t >> 7, k = t & 127;
        Wt2[t] = out_W[k * F + j];             // transpose out_W
    }
}

// ---------------- fp32 WMMA GEMM: C[n, nOut] = A[n,128] @ Wt^T (+bias, relu) --------------
// block = 128 threads = 4 waves; each wave owns a 16-row M tile; loops over 16-col N tiles.
__global__ __launch_bounds__(128) void node_gemm(const float* __restrict__ A,
                                                 const float* __restrict__ Wt,
                                                 const float* __restrict__ bias,
                                                 float* __restrict__ C,
                                                 int nNodes, int nOut, int doRelu) {
    __shared__ float As[64][132];   // +4 pad: bank = (4*row + k) % 64, conflict-free
    __shared__ float Ws[16][132];
    const int tid  = threadIdx.x;
    const int wave = tid >> 5, lane = tid & 31;
    const int half = lane >> 4, lr = lane & 15;
    const int m0   = blockIdx.x * 64;

    // stage A tile (64x128) -> LDS, coalesced float4
    for (int idx = tid; idx < 64 * 32; idx += 128) {
        int row = idx >> 5, c4 = (idx & 31) << 2;
        int g = m0 + row;
        float4 v = make_float4(0.f, 0.f, 0.f, 0.f);
        if (g < nNodes) v = *(const float4*)(A + (size_t)g * F + c4);
        float* p = &As[row][c4];
        p[0] = v.x; p[1] = v.y; p[2] = v.z; p[3] = v.w;
    }

    const int ntMax = nOut >> 4;
    for (int nt = 0; nt < ntMax; ++nt) {
        __syncthreads();
        const int n0 = nt << 4;
        for (int idx = tid; idx < 16 * 32; idx += 128) {   // stage W slice (16x128)
            int row = idx >> 5, c4 = (idx & 31) << 2;
            float4 v = *(const float4*)(Wt + (size_t)(n0 + row) * F + c4);
            float* p = &Ws[row][c4];
            p[0] = v.x; p[1] = v.y; p[2] = v.z; p[3] = v.w;
        }
        __syncthreads();

        v8f acc = {0.f, 0.f, 0.f, 0.f, 0.f, 0.f, 0.f, 0.f};
        const int arow = (wave << 4) + lr;
#pragma unroll
        for (int k = 0; k < F; k += 4) {
            // A 16x4 frag: lanes0-15 {K,K+1}, lanes16-31 {K+2,K+3}  (ISA 7.12.2)
            v2f a = *(const v2f*)&As[arow][k + (half << 1)];
            // B 4x16 frag mirrors A across lanes
            v2f b = *(const v2f*)&Ws[lr][k + (half << 1)];
#if defined(__gfx1250__)
            acc = __builtin_amdgcn_wmma_f32_16x16x4_f32(false, a, false, b,
                                                        (short)0, acc, false, false);
#else
            acc[0] += a[0] * b[0] + a[1] * b[1];   // host-pass placeholder only
#endif
        }
        const int col = n0 + lr;
        const float bv = bias ? bias[col] : 0.f;
#pragma unroll
        for (int r = 0; r < 8; ++r) {             // C: VGPR r -> row r (low) / r+8 (high)
            int node = m0 + (wave << 4) + r + (half << 3);
            if (node < nNodes) {
                float cv = acc[r] + bv;
                if (doRelu) cv = cv > 0.f ? cv : 0.f;
                C[(size_t)node * nOut + col] = cv;
            }
        }
    }
}

// ---------------- edge logits + segment max (wave per edge) ----------------
__global__ __launch_bounds__(256) void edge_logits(const long long* __restrict__ ei,
                                                   const float* __restrict__ P,
                                                   const float* __restrict__ attn_w,
                                                   float* __restrict__ logits,
                                                   float* __restrict__ seg_max) {
    const int lane = threadIdx.x & 31;
    const int e = blockIdx.x * 8 + (threadIdx.x >> 5);
    if (e >= ET) return;
    long long s, d;
    if (e < NE) { s = ei[e]; d = ei[NE + e]; } else { s = d = e - NE; }
    const int c = lane << 2;                      // cols [0,128): head = c/64
    float4 a = *(const float4*)(P + (size_t)s * PW + c);
    float4 b = *(const float4*)(P + (size_t)d * PW + 128 + c);   // bias pre-folded
    float4 w = *(const float4*)(attn_w + c);
    float p = lrelu(a.x + b.x) * w.x + lrelu(a.y + b.y) * w.y +
              lrelu(a.z + b.z) * w.z + lrelu(a.w + b.w) * w.w;
    p += __shfl_xor(p, 1, 16);
    p += __shfl_xor(p, 2, 16);
    p += __shfl_xor(p, 4, 16);
    p += __shfl_xor(p, 8, 16);
    if ((lane & 15) == 0) {
        int h = lane >> 4;
        logits[(size_t)e * 2 + h] = p;
        atomicMaxF(seg_max + (size_t)d * 2 + h, p);
    }
}

// ---------------- exp(logit - max) + segment sum (thread per edge) ----------------
__global__ void edge_expsum(const long long* __restrict__ ei, float* __restrict__ logits,
                            const float* __restrict__ seg_max, float* __restrict__ seg_sum) {
    int e = blockIdx.x * 256 + threadIdx.x;
    if (e >= ET) return;
    long long d = (e < NE) ? ei[NE + e] : (long long)(e - NE);
    float e0 = expf(logits[(size_t)e * 2 + 0] - seg_max[(size_t)d * 2 + 0]);
    float e1 = expf(logits[(size_t)e * 2 + 1] - seg_max[(size_t)d * 2 + 1]);
    logits[(size_t)e * 2 + 0] = e0;               // store ex in place
    logits[(size_t)e * 2 + 1] = e1;
    atomicAdd(seg_sum + (size_t)d * 2 + 0, e0);
    atomicAdd(seg_sum + (size_t)d * 2 + 1, e1);
}

// ---------------- msg = V[src]*alpha scatter-add (wave per edge) ----------------
__global__ __launch_bounds__(256) void edge_aggregate(const long long* __restrict__ ei,
                                                      const float* __restrict__ P,
                                                      const float* __restrict__ logits,
                                                      const float* __restrict__ seg_sum,
                                                      float* __restrict__ agg) {
    const int lane = threadIdx.x & 31;
    const int e = blockIdx.x * 8 + (threadIdx.x >> 5);
    if (e >= ET) return;
    long long s, d;
    if (e < NE) { s = ei[e]; d = ei[NE + e]; } else { s = d = e - NE; }
    const int h = lane >> 4;
    const float alpha = logits[(size_t)e * 2 + h] / seg_sum[(size_t)d * 2 + h];
    const int c = lane << 2;
    float4 v = *(const float4*)(P + (size_t)s * PW + 256 + c);
    float* ag = agg + (size_t)d * F + c;
    atomicAdd(ag + 0, v.x * alpha);
    atomicAdd(ag + 1, v.y * alpha);
    atomicAdd(ag + 2, v.z * alpha);
    atomicAdd(ag + 3, v.w * alpha);
}

// ---------------- phase head + softmax (wave per node) ----------------
__global__ __launch_bounds__(256) void phase_softmax(const float* __restrict__ h2,
                                                     const float* __restrict__ phase_W,
                                                     const float* __restrict__ phase_b,
                                                     float* __restrict__ out, int n) {
    const int lane = threadIdx.x & 31;
    const int node = blockIdx.x * 8 + (threadIdx.x >> 5);
    if (node >= n) return;
    const int c = lane << 2;
    float4 v = *(const float4*)(h2 + (size_t)node * F + c);
    const float4* pw = (const float4*)(phase_W + (size_t)c * 4);  // rows c..c+3 of (128,4)
    float4 r0 = pw[0], r1 = pw[1], r2 = pw[2], r3 = pw[3];
    float4 acc;
    acc.x = v.x * r0.x + v.y * r1.x + v.z * r2.x + v.w * r3.x;
    acc.y = v.x * r0.y + v.y * r1.y + v.z * r2.y + v.w * r3.y;
    acc.z = v.x * r0.z + v.y * r1.z + v.z * r2.z + v.w * r3.z;
    acc.w = v.x * r0.w + v.y * r1.w + v.z * r2.w + v.w * r3.w;
#pragma unroll
    for (int m = 16; m >= 1; m >>= 1) {
        acc.x += __shfl_xor(acc.x, m);
        acc.y += __shfl_xor(acc.y, m);
        acc.z += __shfl_xor(acc.z, m);
        acc.w += __shfl_xor(acc.w, m);
    }
    if (lane == 0) {
        acc.x += phase_b[0]; acc.y += phase_b[1]; acc.z += phase_b[2]; acc.w += phase_b[3];
        float mx = fmaxf(fmaxf(acc.x, acc.y), fmaxf(acc.z, acc.w));
        float e0 = expf(acc.x - mx), e1 = expf(acc.y - mx);
        float e2 = expf(acc.z - mx), e3 = expf(acc.w - mx);
        float inv = 1.f / (e0 + e1 + e2 + e3);
        float4 o = make_float4(e0 * inv, e1 * inv, e2 * inv, e3 * inv);
        *(float4*)(out + (size_t)node * 4) = o;
    }
}

extern "C" void kernel_launch(void* const* d_in, const int* in_sizes, int n_in,
                              void* d_out, int out_size, void* d_ws, size_t ws_size,
                              hipStream_t stream) {
    (void)in_sizes; (void)n_in; (void)out_size; (void)ws_size;
    const float*     h       = (const float*)d_in[0];
    const long long* ei      = (const long long*)d_in[1];   // int64 per reference
    const float*     pair_W  = (const float*)d_in[2];
    const float*     pair_b  = (const float*)d_in[3];
    const float*     attn_w  = (const float*)d_in[4];
    const float*     value_W = (const float*)d_in[5];
    const float*     out_W   = (const float*)d_in[6];
    const float*     out_b   = (const float*)d_in[7];
    const float*     phase_W = (const float*)d_in[8];
    const float*     phase_b = (const float*)d_in[9];

    float* ws      = (float*)d_ws;
    float* P       = ws;                               // [NN, 384]: Asrc|Adst(+b)|V
    float* Wt1     = P + (size_t)NN * PW;              // [384,128]
    float* bias1   = Wt1 + PW * F;                     // [384]
    float* Wt2     = bias1 + PW;                       // [128,128]
    float* logits  = Wt2 + F * F;                      // [ET,2] (then ex in place)
    float* seg_max = logits + (size_t)ET * 2;          // [NN,2]
    float* seg_sum = seg_max + 2 * NN;                 // [NN,2]
    float* agg     = seg_sum + 2 * NN;                 // [NN,128]
    float* h2      = P;                                // alias: P dead after aggregation

    init_ws<<<(F * NN + 255) / 256, 256, 0, stream>>>(seg_max, seg_sum, agg);
    build_weights<<<(PW * F + PW + F * F + 255) / 256, 256, 0, stream>>>(
        pair_W, pair_b, value_W, out_W, Wt1, bias1, Wt2);
    node_gemm<<<(NN + 63) / 64, 128, 0, stream>>>(h, Wt1, bias1, P, NN, PW, 0);
    edge_logits<<<(ET + 7) / 8, 256, 0, stream>>>(ei, P, attn_w, logits, seg_max);
    edge_expsum<<<(ET + 255) / 256, 256, 0, stream>>>(ei, logits, seg_max, seg_sum);
    edge_aggregate<<<(ET + 7) / 8, 256, 0, stream>>>(ei, P, logits, seg_sum, agg);
    node_gemm<<<(NN + 63) / 64, 128, 0, stream>>>(agg, Wt2, out_b, h2, NN, F, 1);
    phase_softmax<<<(NN + 7) / 8, 256, 0, stream>>>(h2, phase_W, phase_b, (float*)d_out, NN);
}